// FCOSDecoder_19645180412512
// MI455X (gfx1250) — compile-verified
//
#include <hip/hip_runtime.h>
#include <math.h>

#define B_IMG 8
#define NCLS 80
#define NTOT 21824          // 16384+4096+1024+256+64
#define MTOT 3320           // 1000+1000+1000+256+64
#define MAXDET 100
#define MIN_SCORE 0.05f
#define NMS_THR 0.6f
#define NEGV -1000000000.0f
#define NPT 4               // ceil(MTOT/1024) elements owned per NMS thread

__constant__ int c_Noff[6] = {0, 16384, 20480, 21504, 21760, 21824};
__constant__ int c_Koff[6] = {0, 1000, 2000, 3000, 3256, 3320};
__constant__ int c_K[5]    = {1000, 1000, 1000, 256, 64};

struct InPtrs { const float* cls[5]; const float* reg[5]; const float* ctr[5]; };

__device__ __forceinline__ float sigmoidf_(float x) { return 1.0f / (1.0f + expf(-x)); }
__device__ __forceinline__ int imin_(int a, int b) { return a < b ? a : b; }
__device__ __forceinline__ int imax_(int a, int b) { return a > b ? a : b; }

// Builtin signature (probed via hipcc diagnostic): v4i pointers, AS1 src, AS3 dst.
typedef int v4i_ __attribute__((vector_size(16)));
typedef __attribute__((address_space(1))) v4i_ as1_v4i;
typedef __attribute__((address_space(3))) v4i_ as3_v4i;

// CDNA5 async global -> LDS copy (16B per lane), tracked by ASYNCcnt.
__device__ __forceinline__ void async_b128(const void* g, void* l) {
#if defined(__HIP_DEVICE_COMPILE__) && defined(__gfx1250__) && __has_builtin(__builtin_amdgcn_global_load_async_to_lds_b128)
  __builtin_amdgcn_global_load_async_to_lds_b128((as1_v4i*)g, (as3_v4i*)l, 0, 0);
#else
  *(float4*)l = *(const float4*)g;
#endif
}

__device__ __forceinline__ void wait_async0() {
#if defined(__HIP_DEVICE_COMPILE__) && defined(__gfx1250__)
#if __has_builtin(__builtin_amdgcn_s_wait_asynccnt)
  __builtin_amdgcn_s_wait_asynccnt(0);
#else
  asm volatile("s_wait_asynccnt 0" ::: "memory");
#endif
#endif
}

// ---------------------------------------------------------------------------
// Kernel 1: per-position score/class. Level is block-uniform (scalar if-chain,
// every per-level segment is a multiple of 256 threads); b/p via shift/mask.
// Memory-bound: 320B cls stream per position; one sigmoid (monotonicity).
// ---------------------------------------------------------------------------
__global__ void fcos_score_kernel(InPtrs P, float* __restrict__ scoresAll,
                                  int* __restrict__ classAll) {
  int blk = blockIdx.x;
  int l, segStart, log2N, noff;
  if (blk < 512)      { l = 0; segStart = 0;   log2N = 14; noff = 0;     }
  else if (blk < 640) { l = 1; segStart = 512; log2N = 12; noff = 16384; }
  else if (blk < 672) { l = 2; segStart = 640; log2N = 10; noff = 20480; }
  else if (blk < 680) { l = 3; segStart = 672; log2N = 8;  noff = 21504; }
  else                { l = 4; segStart = 680; log2N = 6;  noff = 21760; }
  int idx = (blk - segStart) * 256 + threadIdx.x;   // segment is exactly B*Nl
  int b = idx >> log2N;
  int p = idx & ((1 << log2N) - 1);
  int Nl = 1 << log2N;

  const float4* cp = (const float4*)(P.cls[l] + ((size_t)b * Nl + p) * NCLS);
  float mv = -3.0e38f; int mi = 0;
#pragma unroll
  for (int q = 0; q < NCLS / 4; ++q) {
    float4 v = cp[q];                       // global_load_b128 stream
    if (v.x > mv) { mv = v.x; mi = 4 * q + 0; }
    if (v.y > mv) { mv = v.y; mi = 4 * q + 1; }
    if (v.z > mv) { mv = v.z; mi = 4 * q + 2; }
    if (v.w > mv) { mv = v.w; mi = 4 * q + 3; }
  }
  float ct = P.ctr[l][(size_t)b * Nl + p];
  float sc = sqrtf(sigmoidf_(mv) * sigmoidf_(ct));
  int gid = b * NTOT + noff + p;
  scoresAll[gid] = sc;
  classAll[gid]  = mi;
}

// ---------------------------------------------------------------------------
// Parallel inclusive suffix scan of a[0..1023] (ping-pong with b2), then pick
// the unique bin t with S(t) >= K > S(t+1). Executed by the whole block.
// ---------------------------------------------------------------------------
__device__ __forceinline__ void suffix_pick_1024(unsigned* a, unsigned* b2,
                                                 int tid, int nthr, unsigned Kthr,
                                                 int* out_t, int* out_above) {
  unsigned* src = a; unsigned* dst = b2;
  for (int off = 1; off < 1024; off <<= 1) {        // 10 steps
    for (int i = tid; i < 1024; i += nthr) {
      unsigned v = src[i];
      if (i + off < 1024) v += src[i + off];
      dst[i] = v;
    }
    __syncthreads();
    unsigned* t = src; src = dst; dst = t;
  }
  for (int i = tid; i < 1024; i += nthr) {          // unique writer
    unsigned Si = src[i];
    unsigned Sn = (i < 1023) ? src[i + 1] : 0u;
    if (Si >= Kthr && Sn < Kthr) { *out_t = i; *out_above = (int)Sn; }
  }
  __syncthreads();
}

// ---------------------------------------------------------------------------
// Kernel 2: per-(image,level) top-K via two-level histogram threshold
// (20-bit effective score resolution), then gather + box decode (exp, snap).
// 40 blocks x 256 threads.
// ---------------------------------------------------------------------------
__global__ void fcos_topk_kernel(InPtrs P,
                                 const float* __restrict__ scoresAll,
                                 const int* __restrict__ classAll,
                                 float* __restrict__ candS,
                                 float* __restrict__ candC,
                                 float* __restrict__ candB) {
  __shared__ unsigned hist[1024];
  __shared__ unsigned hist2[1024];
  __shared__ int sh_t, sh_above, sh_t2, sh_above2;
  __shared__ unsigned cnt_hi, cnt_mid, cnt_eq;

  int l = blockIdx.x % 5;
  int b = blockIdx.x / 5;
  int wf = 128 >> l;
  int Nl = wf * wf;
  int K  = c_K[l];
  int tid = threadIdx.x;
  int nthr = blockDim.x;
  const float* sBase = scoresAll + (size_t)b * NTOT + c_Noff[l];
  const int*   cBase = classAll  + (size_t)b * NTOT + c_Noff[l];
  bool need_sel = (Nl > K);   // block-uniform

  if (need_sel) {
    for (int i = tid; i < 1024; i += nthr) hist[i] = 0;
    if (tid == 0) { cnt_hi = 0; cnt_mid = 0; cnt_eq = 0; }
    __syncthreads();
    for (int p = tid; p < Nl; p += nthr) {
      int bin = imin_((int)(sBase[p] * 1024.0f), 1023);
      atomicAdd(&hist[bin], 1u);
    }
    __syncthreads();
    suffix_pick_1024(hist, hist2, tid, nthr, (unsigned)K, &sh_t, &sh_above);
    int tt = sh_t;
    // refine inside threshold bin
    for (int i = tid; i < 1024; i += nthr) hist[i] = 0;
    __syncthreads();
    for (int p = tid; p < Nl; p += nthr) {
      float s = sBase[p];
      int bin = imin_((int)(s * 1024.0f), 1023);
      if (bin == tt) {
        int sub = imax_(imin_((int)(s * 1048576.0f) - tt * 1024, 1023), 0);
        atomicAdd(&hist[sub], 1u);
      }
    }
    __syncthreads();
    suffix_pick_1024(hist, hist2, tid, nthr, (unsigned)(K - sh_above), &sh_t2, &sh_above2);
  }

  for (int p = tid; p < Nl; p += nthr) {
    float s = sBase[p];
    int slot;
    if (!need_sel) {
      slot = p;
    } else {
      int bin = imin_((int)(s * 1024.0f), 1023);
      if (bin > sh_t) {
        slot = (int)atomicAdd(&cnt_hi, 1u);
      } else if (bin == sh_t) {
        int sub = imax_(imin_((int)(s * 1048576.0f) - sh_t * 1024, 1023), 0);
        if (sub > sh_t2) {
          slot = sh_above + (int)atomicAdd(&cnt_mid, 1u);
        } else if (sub == sh_t2) {
          slot = sh_above + sh_above2 + (int)atomicAdd(&cnt_eq, 1u);
          if (slot >= K) continue;
        } else continue;
      } else continue;
    }
    // box decode: exp(reg), pos=( (w+.5)s, (h+.5)s ), int-snap, clamp
    int w = p & (wf - 1), h = p >> (7 - l);
    float stride = (float)(8 << l);
    float px = (w + 0.5f) * stride;
    float py = (h + 0.5f) * stride;
    const float* rg = P.reg[l] + ((size_t)b * Nl + p) * 4;
    float e0 = expf(rg[0]), e1 = expf(rg[1]), e2 = expf(rg[2]), e3 = expf(rg[3]);
    int i0 = imax_((int)(px - e0), 0);
    int i1 = imax_((int)(py - e1), 0);
    int i2 = imin_((int)(px + e2), 1023);
    int i3 = imin_((int)(py + e3), 1023);
    int o = b * MTOT + c_Koff[l] + slot;
    candS[o] = s;
    candC[o] = (float)cBase[p];
    ((float4*)candB)[o] = make_float4((float)i0, (float)i1, (float)i2, (float)i3);
  }
}

// ---------------------------------------------------------------------------
// Kernel 3: greedy NMS, one workgroup (32 wave32s) per image. Box table staged
// once into LDS with async global->LDS b128 (reused 100x for the broadcast
// s_bx[bidx] lookup). Scores/own-boxes/areas live in registers (fixed owner
// per element); suppression fuses with next-iteration local argmax.
// ---------------------------------------------------------------------------
__global__ void __launch_bounds__(1024)
fcos_nms_kernel(const float* __restrict__ candS, const float* __restrict__ candC,
                const float* __restrict__ candB, float* __restrict__ out) {
  __shared__ float4 s_bx[MTOT];
  __shared__ float  redv[32];
  __shared__ int    redi[32];
  __shared__ float  sh_bval;
  __shared__ int    sh_bidx;

  int b = blockIdx.x;
  int tid = threadIdx.x;
  const float4* gbx = (const float4*)candB + (size_t)b * MTOT;
  for (int i = tid; i < MTOT; i += 1024) async_b128(&gbx[i], &s_bx[i]);
  wait_async0();
  __syncthreads();

  // private per-thread state: elements i = tid + k*1024
  const float* gsc = candS + (size_t)b * MTOT;
  float  sc[NPT];
  float4 bx[NPT];
  float  ar[NPT];
#pragma unroll
  for (int k = 0; k < NPT; ++k) {
    int i = tid + k * 1024;
    if (i < MTOT) {
      float s = gsc[i];
      sc[k] = (s > MIN_SCORE) ? s : NEGV;
      bx[k] = s_bx[i];
      ar[k] = (bx[k].z - bx[k].x) * (bx[k].w - bx[k].y);
    } else {
      sc[k] = NEGV; bx[k] = make_float4(0.f, 0.f, 0.f, 0.f); ar[k] = 0.f;
    }
  }
  float lbv = -3.0e38f; int lbi = 0x7FFFFFFF;
#pragma unroll
  for (int k = 0; k < NPT; ++k) {
    int i = tid + k * 1024;
    if (sc[k] > lbv || (sc[k] == lbv && i < lbi)) { lbv = sc[k]; lbi = i; }
  }

  float* outS = out;
  float* outC = out + B_IMG * MAXDET;
  float* outB = out + 2 * B_IMG * MAXDET;
  int wid = tid >> 5, lid = tid & 31;

  for (int it = 0; it < MAXDET; ++it) {
    // block argmax (first-occurrence tie-break, matching jnp.argmax)
    float bv = lbv; int bi = lbi;
    for (int off = 16; off > 0; off >>= 1) {
      float ov = __shfl_xor(bv, off, 32);
      int   oi = __shfl_xor(bi, off, 32);
      if (ov > bv || (ov == bv && oi < bi)) { bv = ov; bi = oi; }
    }
    if (lid == 0) { redv[wid] = bv; redi[wid] = bi; }
    __syncthreads();
    if (wid == 0) {
      bv = redv[lid]; bi = redi[lid];
      for (int off = 16; off > 0; off >>= 1) {
        float ov = __shfl_xor(bv, off, 32);
        int   oi = __shfl_xor(bi, off, 32);
        if (ov > bv || (ov == bv && oi < bi)) { bv = ov; bi = oi; }
      }
      if (lid == 0) { sh_bval = bv; sh_bidx = bi; }
    }
    __syncthreads();

    float bval = sh_bval; int bidx = sh_bidx;
    bool ok = bval > MIN_SCORE;
    float4 bb = s_bx[bidx];                 // broadcast lookup from LDS
    float barea = (bb.z - bb.x) * (bb.w - bb.y);
    if (tid == 0) {
      outS[b * MAXDET + it] = ok ? bval : -1.0f;
      outC[b * MAXDET + it] = ok ? candC[(size_t)b * MTOT + bidx] : -1.0f;
      float4 ob = ok ? bb : make_float4(-1.f, -1.f, -1.f, -1.f);
      ((float4*)outB)[b * MAXDET + it] = ob;
    }
    // suppress + compute next iteration's local best (fused)
    lbv = -3.0e38f; lbi = 0x7FFFFFFF;
#pragma unroll
    for (int k = 0; k < NPT; ++k) {
      int i = tid + k * 1024;
      if (i == bidx) { sc[k] = NEGV; }
      else if (ok && sc[k] > NEGV) {
        float iw = fmaxf(fminf(bx[k].z, bb.z) - fmaxf(bx[k].x, bb.x), 0.0f);
        float ih = fmaxf(fminf(bx[k].w, bb.w) - fmaxf(bx[k].y, bb.y), 0.0f);
        float inter = iw * ih;
        float iou = inter / (ar[k] + barea - inter + 1e-12f);
        if (iou > NMS_THR) sc[k] = NEGV;
      }
      if (sc[k] > lbv || (sc[k] == lbv && i < lbi)) { lbv = sc[k]; lbi = i; }
    }
    __syncthreads();   // protects redv/redi + sh_* reuse next iteration
  }
}

// ---------------------------------------------------------------------------
extern "C" void kernel_launch(void* const* d_in, const int* in_sizes, int n_in,
                              void* d_out, int out_size, void* d_ws, size_t ws_size,
                              hipStream_t stream) {
  (void)in_sizes; (void)n_in; (void)out_size; (void)ws_size;
  // setup_inputs dict order: cls0, reg0, ctr0, cls1, reg1, ctr1, ...
  InPtrs P;
  for (int i = 0; i < 5; ++i) {
    P.cls[i] = (const float*)d_in[3 * i + 0];
    P.reg[i] = (const float*)d_in[3 * i + 1];
    P.ctr[i] = (const float*)d_in[3 * i + 2];
  }
  // workspace layout (~2.0 MB, 16B-aligned sections)
  float* scoresAll = (float*)d_ws;                         // B*NTOT
  int*   classAll  = (int*)(scoresAll + B_IMG * NTOT);     // B*NTOT
  float* candS     = (float*)(classAll + B_IMG * NTOT);    // B*MTOT
  float* candC     = candS + B_IMG * MTOT;                 // B*MTOT
  float* candB     = candC + B_IMG * MTOT;                 // B*MTOT*4

  fcos_score_kernel<<<682, 256, 0, stream>>>(P, scoresAll, classAll);
  fcos_topk_kernel<<<B_IMG * 5, 256, 0, stream>>>(P, scoresAll, classAll,
                                                  candS, candC, candB);
  fcos_nms_kernel<<<B_IMG, 1024, 0, stream>>>(candS, candC, candB, (float*)d_out);
}